// WeightedBinaryCrossEntropyLoss_12386685681700
// MI455X (gfx1250) — compile-verified
//
#include <hip/hip_runtime.h>
#include <hip/hip_bf16.h>

// CDNA5 / gfx1250 WMMA types
typedef __attribute__((ext_vector_type(2))) float v2f;
typedef __attribute__((ext_vector_type(8))) float v8f;

#define NBLOCKS 4096
#define NTHREADS 256
#define EPSW 15.0f

__constant__ float POS_RATIO_C[32] = {
    0.02f, 0.05f, 0.08f, 0.10f, 0.12f, 0.15f, 0.18f, 0.20f,
    0.22f, 0.25f, 0.28f, 0.30f, 0.32f, 0.35f, 0.38f, 0.40f,
    0.42f, 0.45f, 0.48f, 0.50f, 0.52f, 0.55f, 0.58f, 0.60f,
    0.62f, 0.65f, 0.68f, 0.70f, 0.72f, 0.75f, 0.78f, 0.80f};

// numerically stable weighted BCE-with-logits, per element
__device__ __forceinline__ float bce_elem(float x, int t, float wp, float wn) {
    float ax   = __builtin_fabsf(x);
    float soft = __logf(1.0f + __expf(-ax));      // log1p(exp(-|x|)), v_exp/v_log
    float per  = fmaxf(x, 0.0f) + soft - (t ? x : 0.0f);
    float w    = t ? wp : wn;
    return w * per;
}

// Stage 1: streaming partial reduction.  One partial (f32) per block.
__global__ void __launch_bounds__(NTHREADS)
bce_partial_kernel(const float* __restrict__ xg, const int* __restrict__ tg,
                   float* __restrict__ partials, int total_elems) {
    const int tid = blockIdx.x * blockDim.x + threadIdx.x;
    const int nth = gridDim.x * blockDim.x;     // multiple of 8 -> channel quartet is loop-invariant
    const int lane = threadIdx.x & 31;
    const int wid  = threadIdx.x >> 5;

    // per-thread fixed channel quartet -> hoist weight selects out of the loop
    const int c0 = (tid * 4) & 31;
    float p0 = POS_RATIO_C[c0 + 0], p1 = POS_RATIO_C[c0 + 1];
    float p2 = POS_RATIO_C[c0 + 2], p3 = POS_RATIO_C[c0 + 3];
    float wp0 = fminf(1.0f / p0, EPSW), wn0 = fminf(1.0f / (1.0f - p0), EPSW);
    float wp1 = fminf(1.0f / p1, EPSW), wn1 = fminf(1.0f / (1.0f - p1), EPSW);
    float wp2 = fminf(1.0f / p2, EPSW), wn2 = fminf(1.0f / (1.0f - p2), EPSW);
    float wp3 = fminf(1.0f / p3, EPSW), wn3 = fminf(1.0f / (1.0f - p3), EPSW);

    const float4* __restrict__ x4 = (const float4*)xg;
    const int4*   __restrict__ t4 = (const int4*)tg;
    const int quads = total_elems >> 2;          // 16B-vector elements
    const int iters = quads / nth;               // uniform trip count (EXEC all-1s for WMMA)

    v8f acc;                                     // WMMA C/D accumulator (16x16 f32 tile)
    acc[0] = 0.f; acc[1] = 0.f; acc[2] = 0.f; acc[3] = 0.f;
    acc[4] = 0.f; acc[5] = 0.f; acc[6] = 0.f; acc[7] = 0.f;
    v2f bones; bones[0] = 1.0f; bones[1] = 1.0f; // all-ones 4x16 B matrix

    for (int k = 0; k < iters; ++k) {
        int i = tid + k * nth;
        float4 xv = x4[i];
        int4   tv = t4[i];
        float v0 = bce_elem(xv.x, tv.x, wp0, wn0);
        float v1 = bce_elem(xv.y, tv.y, wp1, wn1);
        float v2 = bce_elem(xv.z, tv.z, wp2, wn2);
        float v3 = bce_elem(xv.w, tv.w, wp3, wn3);
        v2f a; a[0] = v0 + v1; a[1] = v2 + v3;   // 16x4 f32 A tile: 2 K-slots per lane
        // C += A x ones : 64 wave-wide adds per instruction on the matrix pipe
        acc = __builtin_amdgcn_wmma_f32_16x16x4_f32(
            /*neg_a=*/false, a, /*neg_b=*/false, bones,
            /*c_mod=*/(short)0, acc, /*reuse_a=*/false, /*reuse_b=*/false);
    }

    // divergent tail (no WMMA under partial EXEC)
    float tail = 0.0f;
    for (int i = iters * nth + tid; i < quads; i += nth) {
        float4 xv = x4[i];
        int4   tv = t4[i];
        tail += bce_elem(xv.x, tv.x, wp0, wn0) + bce_elem(xv.y, tv.y, wp1, wn1) +
                bce_elem(xv.z, tv.z, wp2, wn2) + bce_elem(xv.w, tv.w, wp3, wn3);
    }

    // Collapse D: each row-sum is replicated across 16 columns, so a full
    // 32-lane butterfly of (row-partial + 16*tail) yields 16 * wave_total.
    float r = acc[0] + acc[1] + acc[2] + acc[3] + acc[4] + acc[5] + acc[6] + acc[7];
    r += 16.0f * tail;
    #pragma unroll
    for (int off = 16; off > 0; off >>= 1)
        r += __shfl_xor(r, off, 32);
    r *= 0.0625f;                                // undo 16x replication

    __shared__ float wsum[NTHREADS / 32];
    if (lane == 0) wsum[wid] = r;
    __syncthreads();
    if (threadIdx.x == 0) {
        float b = 0.0f;
        #pragma unroll
        for (int i = 0; i < NTHREADS / 32; ++i) b += wsum[i];
        partials[blockIdx.x] = b;
    }
}

// Stage 2: reduce the 4096 block partials in f64 and write the mean.
__global__ void __launch_bounds__(256)
bce_final_kernel(const float* __restrict__ partials, float* __restrict__ out,
                 int nparts, float inv_total) {
    __shared__ double red[256];
    double s = 0.0;
    for (int i = threadIdx.x; i < nparts; i += 256) s += (double)partials[i];
    red[threadIdx.x] = s;
    __syncthreads();
    for (int off = 128; off > 0; off >>= 1) {
        if (threadIdx.x < off) red[threadIdx.x] += red[threadIdx.x + off];
        __syncthreads();
    }
    if (threadIdx.x == 0) out[0] = (float)(red[0] * (double)inv_total);
}

extern "C" void kernel_launch(void* const* d_in, const int* in_sizes, int n_in,
                              void* d_out, int out_size, void* d_ws, size_t ws_size,
                              hipStream_t stream) {
    const float* x = (const float*)d_in[0];
    const int*   t = (const int*)d_in[1];
    float* out = (float*)d_out;
    float* partials = (float*)d_ws;             // NBLOCKS floats of scratch
    const int total = in_sizes[0];              // N*C = 33554432

    bce_partial_kernel<<<NBLOCKS, NTHREADS, 0, stream>>>(x, t, partials, total);
    bce_final_kernel<<<1, 256, 0, stream>>>(partials, out, NBLOCKS,
                                            1.0f / (float)total);
}